// SoftFireCA_1374389534749
// MI455X (gfx1250) — compile-verified
//
#include <hip/hip_runtime.h>
#include <math.h>

#define GH 2048
#define GW 2048
#define NCELL (GH * GW)
#define N_STEPS 20
#define TX 32
#define TY 8
// Padded LDS tile row: [3] = left halo, [4..35] = interior (16B-aligned start),
// [36] = right halo.  Row stride 40 floats = 160B keeps every interior chunk
// 16B-aligned for b128 async-to-LDS.
#define LW 40
#define LH (TY + 2)          // 10 rows (tile + halo)
#define LN (LW * LH)

// ---------------------------------------------------------------------------
// CDNA5 async global->LDS (ASYNCcnt path).  Low 32 bits of a generic pointer
// to LDS are the group-segment byte offset (HW
// adds the wave's LDS_BASE).  Disabled lanes (EXEC=0) do not issue.
// ---------------------------------------------------------------------------
__device__ __forceinline__ unsigned int lds_off(const void* p) {
    return (unsigned int)(unsigned long long)p;
}
__device__ __forceinline__ void async_ld_b32(const float* gptr, float* lptr) {
    asm volatile("global_load_async_to_lds_b32 %0, %1, off"
                 :: "v"(lds_off(lptr)), "v"(gptr) : "memory");
}
__device__ __forceinline__ void async_ld_b128(const float* gptr, float* lptr) {
    asm volatile("global_load_async_to_lds_b128 %0, %1, off"
                 :: "v"(lds_off(lptr)), "v"(gptr) : "memory");
}
__device__ __forceinline__ void wait_async_all() {
    asm volatile("s_wait_asynccnt 0x0" ::: "memory");
}

// ---------------------------------------------------------------------------
// Hot kernel (20 launches): one CA step.  Stages haloed tiles of state /
// wind / height into LDS via wide async-to-LDS, recomputes slope factor phi
// from the L2-resident height plane (instead of streaming 128MB of
// precomputed coef planes, which would spill the 192MB L2), then does the
// 8-offset FMA accumulation, clips, and updates arrival in place.
// ---------------------------------------------------------------------------
__global__ __launch_bounds__(TX * TY)
void fire_step_kernel(const float* __restrict__ s_in,
                      float* __restrict__ s_out,
                      const float* __restrict__ wind,
                      const float* __restrict__ height,
                      const float* __restrict__ gain,
                      const float* __restrict__ log_gamma,
                      float* __restrict__ arrival,
                      float t) {
    __shared__ __align__(16) float s_t[LN];
    __shared__ __align__(16) float w_t[LN];
    __shared__ __align__(16) float h_t[LN];

    const int gi0 = blockIdx.y * TY;   // tile origin (row)
    const int gj0 = blockIdx.x * TX;   // tile origin (col), multiple of 32
    const int tid = threadIdx.y * TX + threadIdx.x;

    if (tid < LH * 8) {
        // Interior: LH rows x 8 chunks of 4 floats.  Global addr is 128B
        // aligned (gj0 % 32 == 0), LDS dest is 16B aligned -> b128 async DMA.
        const int row   = tid >> 3;
        const int chunk = tid & 7;
        const int gi    = gi0 + row - 1;
        const int l     = row * LW + 4 + chunk * 4;
        if ((unsigned)gi < GH) {
            const size_t g = (size_t)gi * GW + (size_t)(gj0 + chunk * 4);
            async_ld_b128(s_in   + g, &s_t[l]);
            async_ld_b128(wind   + g, &w_t[l]);
            async_ld_b128(height + g, &h_t[l]);
        } else {
            const float4 z = make_float4(0.f, 0.f, 0.f, 0.f);
            *(float4*)&s_t[l] = z;
            *(float4*)&w_t[l] = z;
            *(float4*)&h_t[l] = z;
        }
    } else if (tid < LH * 8 + 2 * LH) {
        // Halo columns: LH rows x {left, right} single cells (corners incl.).
        const int k    = tid - LH * 8;
        const int row  = k % LH;
        const int side = k / LH;                     // 0 = left, 1 = right
        const int gi   = gi0 + row - 1;
        const int gj   = side ? (gj0 + TX) : (gj0 - 1);
        const int l    = row * LW + (side ? 36 : 3);
        if ((unsigned)gi < GH && (unsigned)gj < GW) {
            const size_t g = (size_t)gi * GW + (size_t)gj;
            async_ld_b32(s_in   + g, &s_t[l]);
            async_ld_b32(wind   + g, &w_t[l]);
            async_ld_b32(height + g, &h_t[l]);
        } else {
            s_t[l] = 0.0f;   // zero halo => zero-padded shift semantics
            w_t[l] = 0.0f;
            h_t[l] = 0.0f;
        }
    }
    wait_async_all();     // drain this wave's ASYNCcnt
    __syncthreads();      // make all waves' LDS writes visible

    const int ti = threadIdx.y;
    const int tj = threadIdx.x;
    const int c  = (ti + 1) * LW + (tj + 4);

    const float gam = __expf(log_gamma[0]);   // uniform scalar (v_s_exp_f32)
    const float h_c = h_t[c];

    float total = 0.0f;
    // phi = exp(gamma*dh) if dh<=0 (downhill), else 1 + gamma*sqrt(dh)
    #define FIRE_ACC(DI, DJ, DIST)                                          \
    {                                                                       \
        const int n = c + (DI) * LW + (DJ);                                 \
        float wsnb = s_t[n] * w_t[n];                                       \
        float dh   = h_c - h_t[n];                                          \
        float down = __expf(gam * dh);                                      \
        float up   = fmaf(gam, __fsqrt_rn(dh > 0.0f ? dh : 0.0f), 1.0f);    \
        float phi  = (dh <= 0.0f) ? down : up;                              \
        total = fmaf((DIST) * phi, wsnb, total);                            \
    }
    FIRE_ACC(-1, -1, 0.83f) FIRE_ACC(-1, 0, 1.00f) FIRE_ACC(-1, 1, 0.83f)
    FIRE_ACC( 0, -1, 1.00f)                        FIRE_ACC( 0, 1, 1.00f)
    FIRE_ACC( 1, -1, 0.83f) FIRE_ACC( 1, 0, 1.00f) FIRE_ACC( 1, 1, 0.83f)
    #undef FIRE_ACC

    const size_t g = (size_t)(gi0 + ti) * GW + (size_t)(gj0 + tj);
    float s_old = s_t[c];
    float s_new = fmaf(gain[g], total, s_old);
    s_new = fminf(fmaxf(s_new, 0.0f), 1.0f);
    s_out[g] = s_new;

    float first_ign = fmaxf(s_new - s_old, 0.0f);
    arrival[g] = arrival[g] - first_ign * ((float)N_STEPS - t);
}

// ---------------------------------------------------------------------------
// One-shot gain plane.  gain = age_factor(age) * exp(-beta*moisture)
//   age_factor = (1+P_MAX)^((age/T_MAX)^alpha) - 1  (P_MAX=1 -> exp2), sat 1
// ---------------------------------------------------------------------------
__global__ void fire_gain_kernel(const float* __restrict__ log_alpha,
                                 const float* __restrict__ log_beta,
                                 const float* __restrict__ age,
                                 const float* __restrict__ moisture,
                                 float* __restrict__ gain) {
    size_t i = (size_t)blockIdx.x * blockDim.x + threadIdx.x;
    if (i >= NCELL) return;
    float alpha = __expf(log_alpha[0]);
    float beta  = __expf(log_beta[0]);
    float a     = age[i];
    float ratio = a * (1.0f / 30.0f);                 // T_MAX = 30
    float below = exp2f(__powf(ratio, alpha)) - 1.0f; // (1+P_MAX)^x - 1
    float af    = (a < 30.0f) ? below : 1.0f;         // saturate at P_MAX = 1
    float psi   = __expf(-beta * moisture[i]);
    gain[i] = af * psi;
}

// ---------------------------------------------------------------------------
// Init.  state0 = delta at (i0,j0)*ignition; arrival = n_steps.  Race-free:
// every thread computes whether it owns the ignition cell.
// ---------------------------------------------------------------------------
__global__ void fire_init_kernel(float* __restrict__ state0,
                                 float* __restrict__ arrival,
                                 const int* __restrict__ i0,
                                 const int* __restrict__ j0,
                                 const float* __restrict__ ignition) {
    size_t i = (size_t)blockIdx.x * blockDim.x + threadIdx.x;
    if (i >= NCELL) return;
    size_t target = (size_t)i0[0] * GW + (size_t)j0[0];
    state0[i]  = (i == target) ? ignition[0] : 0.0f;
    arrival[i] = (float)N_STEPS;
}

// ---------------------------------------------------------------------------
// Host launcher.  n_steps=20 / n_substeps=1 are fixed by setup_inputs (the
// device-resident scalars cannot drive host launch counts under graph
// capture).  Workspace: gain | stateA | stateB = 3 * 16MB = 48MB.
// ---------------------------------------------------------------------------
extern "C" void kernel_launch(void* const* d_in, const int* in_sizes, int n_in,
                              void* d_out, int out_size, void* d_ws, size_t ws_size,
                              hipStream_t stream) {
    (void)in_sizes; (void)n_in; (void)out_size; (void)ws_size;

    const float* log_alpha = (const float*)d_in[0];
    const float* log_beta  = (const float*)d_in[1];
    const float* log_gamma = (const float*)d_in[2];
    const float* height    = (const float*)d_in[3];
    const float* age       = (const float*)d_in[4];
    const float* moisture  = (const float*)d_in[5];
    const float* wind      = (const float*)d_in[6];
    const float* ignition  = (const float*)d_in[7];
    const int*   i0        = (const int*)d_in[8];
    const int*   j0        = (const int*)d_in[9];

    float* arrival = (float*)d_out;
    float* gain    = (float*)d_ws;
    float* stateA  = gain   + NCELL;
    float* stateB  = stateA + NCELL;

    dim3 b1(256), g1((NCELL + 255) / 256);
    fire_gain_kernel<<<g1, b1, 0, stream>>>(log_alpha, log_beta, age, moisture, gain);
    fire_init_kernel<<<g1, b1, 0, stream>>>(stateA, arrival, i0, j0, ignition);

    dim3 bs(TX, TY), gs(GW / TX, GH / TY);
    float* cur = stateA;
    float* nxt = stateB;
    for (int t = 1; t <= N_STEPS; ++t) {
        fire_step_kernel<<<gs, bs, 0, stream>>>(cur, nxt, wind, height, gain,
                                                log_gamma, arrival, (float)t);
        float* tmp = cur; cur = nxt; nxt = tmp;
    }
}